// SinkhornLoss_6459630813760
// MI455X (gfx1250) — compile-verified
//
#include <hip/hip_runtime.h>
#include <stdint.h>

#define N_ITERS 20
#define NEG_INF (-INFINITY)

typedef float v2f __attribute__((ext_vector_type(2)));
typedef float v8f __attribute__((ext_vector_type(8)));

// ---------------------------------------------------------------- init
__global__ __launch_bounds__(256)
void sk_init(float* __restrict__ g, float* __restrict__ out, int BN) {
    int idx = blockIdx.x * blockDim.x + threadIdx.x;
    if (idx < BN) g[idx] = 0.0f;
    if (idx == 0) out[0] = 0.0f;
}

// ---------------------------------------------------------------- n_real per batch
__global__ __launch_bounds__(256)
void sk_nreal(const uint8_t* __restrict__ mask, float* __restrict__ nreal, int N) {
    __shared__ float red[256];
    int b = blockIdx.x;
    float c = 0.0f;
    for (int j = threadIdx.x; j < N; j += 256)
        c += (mask[(long long)b * N + j] ? 1.0f : 0.0f);
    red[threadIdx.x] = c;
    __syncthreads();
    for (int o = 128; o > 0; o >>= 1) {
        if (threadIdx.x < o) red[threadIdx.x] += red[threadIdx.x + o];
        __syncthreads();
    }
    if (threadIdx.x == 0) nreal[b] = fmaxf(red[0], 1.0f);
}

// ---------------------------------------------------------------- row step: f_i = -LSE_j(-D/eps + g_j)
// One wave per row. D row streamed through LDS with double-buffered
// global_load_async_to_lds_b128 (ASYNCcnt pipeline), branchless online LSE
// per lane (mask folded in as a -inf bias), wave32 shuffle merge.
__global__ __launch_bounds__(256)
void sk_row(const float* __restrict__ D, const uint8_t* __restrict__ mask,
            const float* __restrict__ g, float* __restrict__ f,
            int B, int N, float nie /* = -1/eps */) {
    __shared__ float buf[2][8][128];   // 2 buffers x 8 waves x 128 floats = 8KB
    const int wave = threadIdx.x >> 5;
    const int lane = threadIdx.x & 31;
    const long long rowIdx = (long long)blockIdx.x * 8 + wave;
    if (rowIdx >= (long long)B * N) return;   // wave-uniform
    const int b = (int)(rowIdx / N);
    const int r = (int)(rowIdx % N);
    const float*   rowD  = D    + rowIdx * (long long)N;
    const float*   grow  = g    + (long long)b * N;
    const uint8_t* cmask = mask + (long long)b * N;
    const bool rowvalid = (cmask[r] != 0);

    const int CH = N >> 7;          // chunks of 128 floats (32 lanes x float4)
    float m = NEG_INF, s = 0.0f;

    // prologue: issue chunk 0
    {
        uint32_t lo = (uint32_t)(size_t)(void*)&buf[0][wave][lane * 4];
        const float* gp = rowD + lane * 4;
        asm volatile("global_load_async_to_lds_b128 %0, %1, off"
                     :: "v"(lo), "v"(gp) : "memory");
    }
    for (int k = 0; k < CH; ++k) {
        if (k + 1 < CH) {
            // buffer (k+1)&1 was consumed at iter k-1; make sure DS reads drained
            asm volatile("s_wait_dscnt 0" ::: "memory");
            uint32_t lo = (uint32_t)(size_t)(void*)&buf[(k + 1) & 1][wave][lane * 4];
            const float* gp = rowD + (long long)(k + 1) * 128 + lane * 4;
            asm volatile("global_load_async_to_lds_b128 %0, %1, off"
                         :: "v"(lo), "v"(gp) : "memory");
            asm volatile("s_wait_asynccnt 1" ::: "memory");   // chunk k complete
        } else {
            asm volatile("s_wait_asynccnt 0" ::: "memory");
        }
        const int j = k * 128 + lane * 4;
        float4   d4 = *(const float4*)&buf[k & 1][wave][lane * 4];
        float4   g4 = *(const float4*)&grow[j];
        uint32_t m4 = *(const uint32_t*)&cmask[j];

        // branchless: masked entries get a -inf bias -> exp() == 0
        float b0 = ((m4 & 0x000000FFu) != 0) ? 0.0f : NEG_INF;
        float b1 = ((m4 & 0x0000FF00u) != 0) ? 0.0f : NEG_INF;
        float b2 = ((m4 & 0x00FF0000u) != 0) ? 0.0f : NEG_INF;
        float b3 = ((m4 & 0xFF000000u) != 0) ? 0.0f : NEG_INF;
        float x0 = fmaf(d4.x, nie, g4.x) + b0;
        float x1 = fmaf(d4.y, nie, g4.y) + b1;
        float x2 = fmaf(d4.z, nie, g4.z) + b2;
        float x3 = fmaf(d4.w, nie, g4.w) + b3;

        float cmx = fmaxf(fmaxf(x0, x1), fmaxf(x2, x3));
        float nm  = fmaxf(m, cmx);
        if (nm > NEG_INF) {
            s = s * __expf(m - nm)
              + __expf(x0 - nm) + __expf(x1 - nm)
              + __expf(x2 - nm) + __expf(x3 - nm);
            m = nm;
        }
    }
    // wave32 merge of (m, s)
    for (int off = 16; off > 0; off >>= 1) {
        float om = __shfl_xor(m, off);
        float os = __shfl_xor(s, off);
        float nm = fmaxf(m, om);
        float a  = (m  > NEG_INF) ? s  * __expf(m  - nm) : 0.0f;
        float c  = (om > NEG_INF) ? os * __expf(om - nm) : 0.0f;
        s = a + c;
        m = nm;
    }
    if (lane == 0)
        f[rowIdx] = (rowvalid && s > 0.0f) ? -(m + __logf(s)) : 0.0f;
}

// ---------------------------------------------------------------- col step partials
// Thread-per-column (coalesced), rows split into RB blocks for occupancy.
__global__ __launch_bounds__(256)
void sk_col(const float* __restrict__ D, const uint8_t* __restrict__ mask,
            const float* __restrict__ f, float* __restrict__ pm, float* __restrict__ ps,
            int B, int N, int RB, float nie) {
    const int j    = blockIdx.x * 256 + threadIdx.x;
    const int rblk = blockIdx.y;
    const int b    = blockIdx.z;
    const int rows = N / RB;
    const int r0   = rblk * rows;
    const float*   Db = D + ((long long)b * N + r0) * (long long)N;
    const float*   fb = f + (long long)b * N;
    const uint8_t* rm = mask + (long long)b * N;
    const float colb = (rm[j] != 0) ? 0.0f : NEG_INF;

    float m = NEG_INF, s = 0.0f;
    for (int i = 0; i < rows; i += 4) {
        if (i + 8 < rows)   // WGP-scope prefetch of the strided column walk
            __builtin_prefetch(&Db[(long long)(i + 8) * N + j], 0, 3);
        float x[4];
#pragma unroll
        for (int u = 0; u < 4; ++u) {
            const int ri = r0 + i + u;
            float d  = Db[(long long)(i + u) * N + j];
            float rb = (rm[ri] != 0) ? colb : NEG_INF;   // branchless mask bias
            x[u] = fmaf(d, nie, fb[ri]) + rb;
        }
        float cmx = fmaxf(fmaxf(x[0], x[1]), fmaxf(x[2], x[3]));
        float nm  = fmaxf(m, cmx);
        if (nm > NEG_INF) {
            s = s * __expf(m - nm)
              + __expf(x[0] - nm) + __expf(x[1] - nm)
              + __expf(x[2] - nm) + __expf(x[3] - nm);
            m = nm;
        }
    }
    const long long o = (((long long)b * N + j) * RB) + rblk;
    pm[o] = m;
    ps[o] = s;
}

// ---------------------------------------------------------------- merge col partials -> g
__global__ __launch_bounds__(256)
void sk_colmerge(const float* __restrict__ pm, const float* __restrict__ ps,
                 float* __restrict__ g, int BN, int RB) {
    int idx = blockIdx.x * blockDim.x + threadIdx.x;
    if (idx >= BN) return;
    float m = NEG_INF, s = 0.0f;
    for (int rb = 0; rb < RB; ++rb) {
        float om = pm[(long long)idx * RB + rb];
        float os = ps[(long long)idx * RB + rb];
        float nm = fmaxf(m, om);
        if (nm > NEG_INF) {
            float a = (m  > NEG_INF) ? s  * __expf(m  - nm) : 0.0f;
            float c = (om > NEG_INF) ? os * __expf(om - nm) : 0.0f;
            s = a + c;
            m = nm;
        }
    }
    g[idx] = (s > 0.0f) ? -(m + __logf(s)) : 0.0f;
}

// ---------------------------------------------------------------- final: sum P*D via WMMA accumulation
// Wave per row. Each lane produces 2 values of exp(-D/eps+f+g)*D per step;
// v_wmma_f32_16x16x4_f32 with B=ones accumulates 64 values/instruction into C.
// Masking is folded into the exp argument as a -inf bias so the loop is fully
// branchless (EXEC stays all-ones around every WMMA, loads stay vectorized).
__global__ __launch_bounds__(256)
void sk_final(const float* __restrict__ D, const uint8_t* __restrict__ mask,
              const float* __restrict__ f, const float* __restrict__ g,
              const float* __restrict__ nreal, float* __restrict__ out,
              int B, int N, float nie) {
    const int wave = threadIdx.x >> 5;
    const int lane = threadIdx.x & 31;
    const long long rowIdx = (long long)blockIdx.x * 8 + wave;
    if (rowIdx >= (long long)B * N) return;   // wave-uniform
    const int b = (int)(rowIdx / N);
    const int r = (int)(rowIdx % N);
    const float*   rowD = D + rowIdx * (long long)N;
    const float*   gb   = g + (long long)b * N;
    const uint8_t* cm   = mask + (long long)b * N;
    const float rowbias = (cm[r] != 0) ? 0.0f : NEG_INF;
    const float fr = f[rowIdx];

    v2f ones; ones.x = 1.0f; ones.y = 1.0f;
    v8f c = {};
    const int K = N >> 6;                     // 64 elements per wmma step
    for (int k = 0; k < K; ++k) {
        const int j = k * 64 + lane * 2;
        float2   d2 = *(const float2*)&rowD[j];
        float2   g2 = *(const float2*)&gb[j];
        uint16_t mm = *(const uint16_t*)&cm[j];
        float b0 = ((mm & 0x00FFu) != 0) ? rowbias : NEG_INF;
        float b1 = ((mm & 0xFF00u) != 0) ? rowbias : NEG_INF;
        v2f a;
        a.x = __expf(fmaf(d2.x, nie, fr + g2.x) + b0) * d2.x;  // exp(-inf)=0 masks out
        a.y = __expf(fmaf(d2.y, nie, fr + g2.y) + b1) * d2.y;
        c = __builtin_amdgcn_wmma_f32_16x16x4_f32(false, a, false, ones,
                                                  (short)0, c, false, false);
    }
    float cs = c[0] + c[1] + c[2] + c[3] + c[4] + c[5] + c[6] + c[7];
    cs += __shfl_xor(cs, 16);                 // grand total in every lane
    if (lane == 0)
        atomicAdd(out, cs / (nreal[b] * (float)B));
}

// ---------------------------------------------------------------- launcher
extern "C" void kernel_launch(void* const* d_in, const int* in_sizes, int n_in,
                              void* d_out, int out_size, void* d_ws, size_t ws_size,
                              hipStream_t stream) {
    const float*   D    = (const float*)d_in[0];
    const uint8_t* mask = (const uint8_t*)d_in[1];
    float*         out  = (float*)d_out;

    const int BN = in_sizes[1];                       // B*N
    const long long total = (long long)in_sizes[0];   // B*N*N
    const int N = (int)(total / BN);
    const int B = BN / N;
    const int RB = 8;
    const float nie = -1.0f / 0.1f;                   // -1/eps

    // workspace layout (floats): f | g | pm | ps | nreal
    float* f     = (float*)d_ws;
    float* g     = f  + BN;
    float* pm    = g  + BN;
    float* ps    = pm + (size_t)BN * RB;
    float* nreal = ps + (size_t)BN * RB;

    sk_init <<<(BN + 255) / 256, 256, 0, stream>>>(g, out, BN);
    sk_nreal<<<B, 256, 0, stream>>>(mask, nreal, N);

    const int rowBlocks = (BN + 7) / 8;
    dim3 colGrid(N / 256, RB, B);
    for (int it = 0; it < N_ITERS; ++it) {
        sk_row     <<<rowBlocks, 256, 0, stream>>>(D, mask, g, f, B, N, nie);
        sk_col     <<<colGrid,   256, 0, stream>>>(D, mask, f, pm, ps, B, N, RB, nie);
        sk_colmerge<<<(BN + 255) / 256, 256, 0, stream>>>(pm, ps, g, BN, RB);
    }
    sk_final<<<rowBlocks, 256, 0, stream>>>(D, mask, f, g, nreal, out, B, N, nie);
}